// LAMLoss_5102421147747
// MI455X (gfx1250) — compile-verified
//
#include <hip/hip_runtime.h>
#include <hip/hip_bf16.h>
#include <stdint.h>

// ---------------- problem constants (match reference) ----------------
#define B_IMG   4
#define C_IN    4
#define H_IMG   512
#define W_PIX   512
#define HW      (H_IMG * W_PIX)        // 262144
#define NPIX    (B_IMG * HW)           // 1048576
#define HALF_N  (NPIX / 2)             // 524288
#define OUT_CH  256
#define P_NUM   2048
#define N_NUM   2048
#define T_PTS   (P_NUM + N_NUM)        // 4096
#define T16     (T_PTS / 16)           // 256
#define NBIN    16384
#define CAND_MAX 8192
#define TAU_INV 10.0f
#define MARGIN  0.5f
#define EPS_F   1e-8f

// Gram super-tiling
#define NS      32                     // 4096 / 128 super-tiles per dim
#define NPAIRS  (NS * (NS + 1) / 2)    // 528 upper-tri super-tile pairs
#define KCH     32                     // K-chunk staged per TDM load
#define LDS_STRIDE 36                  // 32 data + 4 pad dwords (bank-conflict-free)
#define PANEL_DW  (128 * LDS_STRIDE)   // dwords per staged panel
#define PANEL_BYTES (PANEL_DW * 4)     // 18432 B
#define NPART   (NPAIRS * 8)           // per-wave partial slots

typedef float v2f __attribute__((ext_vector_type(2)));
typedef float v8f __attribute__((ext_vector_type(8)));
typedef unsigned int u32;
typedef unsigned long long u64;
typedef unsigned int v4u __attribute__((ext_vector_type(4)));
typedef int v4i __attribute__((ext_vector_type(4)));
typedef int v8i __attribute__((ext_vector_type(8)));

// ---------------- workspace layout (bytes) ----------------
#define OFF_SCOREP   ((size_t)0)
#define OFF_SCOREN   ((size_t)(OFF_SCOREP + 4ull*NPIX))
#define OFF_ZERO     ((size_t)(OFF_SCOREN + 4ull*NPIX))
#define OFF_HISTP    OFF_ZERO
#define OFF_HISTN    ((size_t)(OFF_HISTP + 4ull*NBIN))
#define OFF_CTRS     ((size_t)(OFF_HISTN + 4ull*NBIN))    // u32[64]
#define OFF_PART     ((size_t)(OFF_CTRS + 256))           // float[NPART]
#define ZERO_BYTES   ((size_t)(4ull*NBIN + 4ull*NBIN + 256 + 4ull*NPART))
#define OFF_THR      ((size_t)(OFF_PART + 4ull*NPART + 224))
#define OFF_SELP     ((size_t)(OFF_THR + 256))
#define OFF_SELN     ((size_t)(OFF_SELP + 4ull*P_NUM))
#define OFF_CANDP    ((size_t)(OFF_SELN + 4ull*N_NUM))
#define OFF_CANDN    ((size_t)(OFF_CANDP + 8ull*CAND_MAX))
#define OFF_NORMED   ((size_t)(OFF_CANDN + 8ull*CAND_MAX))

// counters: [0]=fillAboveP [1]=candCntP [2]=fillAboveN [3]=candCntN [4]=nnz

// ---------------- threefry2x32 (exact JAX PRNG) ----------------
__device__ __forceinline__ void tf2x32(u32 k0, u32 k1, u32 x0, u32 x1,
                                       u32& o0, u32& o1) {
  u32 k2 = 0x1BD11BDAu ^ k0 ^ k1;
  x0 += k0; x1 += k1;
#define TFR(r) { x0 += x1; x1 = (x1 << (r)) | (x1 >> (32 - (r))); x1 ^= x0; }
  TFR(13) TFR(15) TFR(26) TFR(6)   x0 += k1; x1 += k2 + 1u;
  TFR(17) TFR(29) TFR(16) TFR(24)  x0 += k2; x1 += k0 + 2u;
  TFR(13) TFR(15) TFR(26) TFR(6)   x0 += k0; x1 += k1 + 3u;
  TFR(17) TFR(29) TFR(16) TFR(24)  x0 += k1; x1 += k2 + 4u;
  TFR(13) TFR(15) TFR(26) TFR(6)   x0 += k2; x1 += k0 + 5u;
#undef TFR
  o0 = x0; o1 = x1;
}

__device__ __forceinline__ void pixel_masks(const float* __restrict__ pfegc,
                                            const float* __restrict__ ppre,
                                            const float* __restrict__ gt,
                                            int i, bool& pos, bool& neg) {
  bool fg = pfegc[i] >= 0.5f;
  int b = i / HW, hw = i - b * HW;
  int base = b * 3 * HW + hw;
  float p0 = ppre[base], p1 = ppre[base + HW], p2 = ppre[base + 2 * HW];
  bool cls0 = (p0 >= p1) && (p0 >= p2);        // argmax==0 (first-occurrence)
  bool pre = fg && cls0;
  bool g = gt[i] > 0.5f;
  pos = pre && g;
  neg = pre && !g;
}

// ---------------- kernel 1: masks + threefry scores + histograms ----------------
__global__ void score_kernel(const float* __restrict__ pfegc,
                             const float* __restrict__ ppre,
                             const float* __restrict__ gt,
                             u32* __restrict__ scoreP, u32* __restrict__ scoreN,
                             u32* __restrict__ histP, u32* __restrict__ histN) {
  int j = blockIdx.x * blockDim.x + threadIdx.x;
  if (j >= HALF_N) return;
  u32 a0, a1, b0, b1;
  tf2x32(0u, 42u, 0u, 2u, a0, a1);     // jax.random.split(key(42), 2)
  tf2x32(0u, 42u, 1u, 3u, b0, b1);
  u32 kp0 = a0, kp1 = b0, kn0 = a1, kn1 = b1;
  u32 up0, up1, un0, un1;
  tf2x32(kp0, kp1, (u32)j, (u32)(j + HALF_N), up0, up1);
  tf2x32(kn0, kn1, (u32)j, (u32)(j + HALF_N), un0, un1);
  int idx[2] = { j, j + HALF_N };
  u32 bp[2] = { up0, up1 };
  u32 bn[2] = { un0, un1 };
  for (int t = 0; t < 2; ++t) {
    int i = idx[t];
    bool pos, neg;
    pixel_masks(pfegc, ppre, gt, i, pos, neg);
    u32 kP = bp[t] >> 9;               // 23-bit key; monotonic with jax uniform
    u32 kN = bn[t] >> 9;
    scoreP[i] = pos ? kP : 0xFFFFFFFFu;
    scoreN[i] = neg ? kN : 0xFFFFFFFFu;
    if (pos) atomicAdd(&histP[kP >> 9], 1u);
    if (neg) atomicAdd(&histN[kN >> 9], 1u);
  }
}

// ---------------- kernel 2: threshold scan ----------------
__global__ void threshold_kernel(const u32* __restrict__ histP,
                                 const u32* __restrict__ histN,
                                 u32* __restrict__ thr) {
  if (threadIdx.x != 0 || blockIdx.x != 0) return;
  const u32* h[2] = { histP, histN };
  const u32 n[2] = { P_NUM, N_NUM };
  for (int s = 0; s < 2; ++s) {
    u32 cum = 0, tb = 0, r = n[s], above = 0;
    for (int b = NBIN - 1; b >= 0; --b) {
      u32 c = h[s][b];
      if (cum + c >= n[s]) { tb = (u32)b; r = n[s] - cum; above = cum; break; }
      cum += c;
      if (b == 0) { tb = 0; r = n[s] - cum; above = cum; }
    }
    thr[s * 3 + 0] = tb; thr[s * 3 + 1] = r; thr[s * 3 + 2] = above;
  }
}

// ---------------- kernel 3: compaction ----------------
__global__ void compact_kernel(const u32* __restrict__ scoreP,
                               const u32* __restrict__ scoreN,
                               const u32* __restrict__ thr,
                               u32* __restrict__ ctrs,
                               u32* __restrict__ selP, u32* __restrict__ selN,
                               u64* __restrict__ candP, u64* __restrict__ candN) {
  int i = blockIdx.x * blockDim.x + threadIdx.x;
  if (i >= NPIX) return;
  u32 sp = scoreP[i];
  if (sp != 0xFFFFFFFFu) {
    u32 bin = sp >> 9;
    if (bin > thr[0]) {
      u32 p = atomicAdd(&ctrs[0], 1u);
      if (p < P_NUM) selP[p] = (u32)i;
    } else if (bin == thr[0]) {
      u32 c = atomicAdd(&ctrs[1], 1u);
      if (c < CAND_MAX) candP[c] = ((u64)sp << 20) | (u64)(0xFFFFFu - (u32)i);
    }
  }
  u32 sn = scoreN[i];
  if (sn != 0xFFFFFFFFu) {
    u32 bin = sn >> 9;
    if (bin > thr[3]) {
      u32 p = atomicAdd(&ctrs[2], 1u);
      if (p < N_NUM) selN[p] = (u32)i;
    } else if (bin == thr[3]) {
      u32 c = atomicAdd(&ctrs[3], 1u);
      if (c < CAND_MAX) candN[c] = ((u64)sn << 20) | (u64)(0xFFFFFu - (u32)i);
    }
  }
}

// ---------------- kernel 4: boundary-bin exact top-r ----------------
__global__ void select_kernel(const u32* __restrict__ thr,
                              const u32* __restrict__ ctrs,
                              const u64* __restrict__ candP,
                              const u64* __restrict__ candN,
                              u32* __restrict__ selP, u32* __restrict__ selN) {
  for (int s = 0; s < 2; ++s) {
    const u64* cand = s == 0 ? candP : candN;
    u32* sel = s == 0 ? selP : selN;
    u32 m = min(ctrs[s == 0 ? 1 : 3], (u32)CAND_MAX);
    u32 r = thr[s * 3 + 1];
    u32 above = thr[s * 3 + 2];
    for (u32 c = threadIdx.x; c < m; c += blockDim.x) {
      u64 key = cand[c];
      u32 rank = 0;
      for (u32 k = 0; k < m; ++k) rank += (cand[k] > key) ? 1u : 0u;
      if (rank < r) sel[above + rank] = 0xFFFFFu - (u32)(key & 0xFFFFFu);
    }
  }
}

// ---------------- kernel 5: gather + Linear(4->256) + L2 normalize ----------------
__global__ void project_kernel(const float* __restrict__ d1,
                               const float* __restrict__ Wfc,
                               const float* __restrict__ bfc,
                               const u32* __restrict__ selP,
                               const u32* __restrict__ selN,
                               float* __restrict__ normed) {
  __shared__ float red[256];
  int p = blockIdx.x;
  int o = threadIdx.x;
  u32 idx = (p < P_NUM) ? selP[p] : selN[p - P_NUM];
  int b = idx / HW, hw = idx - b * HW;
  float f0 = d1[(b * 4 + 0) * HW + hw];
  float f1 = d1[(b * 4 + 1) * HW + hw];
  float f2 = d1[(b * 4 + 2) * HW + hw];
  float f3 = d1[(b * 4 + 3) * HW + hw];
  float proj = bfc[o] + f0 * Wfc[o * 4 + 0] + f1 * Wfc[o * 4 + 1]
                      + f2 * Wfc[o * 4 + 2] + f3 * Wfc[o * 4 + 3];
  red[o] = proj * proj;
  __syncthreads();
  for (int s = 128; s > 0; s >>= 1) {
    if (o < s) red[o] += red[o + s];
    __syncthreads();
  }
  float nrm = fmaxf(sqrtf(red[0]), EPS_F);
  normed[p * OUT_CH + o] = proj / nrm;
}

// -------- TDM: stage a 128-row x 32-K f32 tile of `normed` into LDS --------
// D# per CDNA5 ISA: group0 = {count, lds_addr, global_addr, type=2};
// group1 = {data_size=4B, pad_enable, pad_interval=128B(code 4),
//           pad_amount=4dw(code 3), dims/strides}. Groups 2/3 zero (2D tensor).
__device__ __forceinline__ void tdm_stage_panel(const float* gbase,
                                                int row0, int kc,
                                                u32 lds_byte_off) {
  u64 gaddr = (u64)(uintptr_t)gbase + ((u64)row0 * OUT_CH + (u64)kc) * 4ull;
  v4u g0;
  g0[0] = 1u;                                   // count=1, user descriptor
  g0[1] = lds_byte_off;                         // lds_addr [63:32]
  g0[2] = (u32)(gaddr & 0xFFFFFFFFull);         // global_addr [95:64]
  g0[3] = (u32)((gaddr >> 32) & 0x01FFFFFFull)  // global_addr [120:96]
          | (2u << 30);                         // type=2 ("image")
  v8i g1;
  g1[0] = (int)((2u << 16)                      // data_size = 4 bytes
              | (1u << 20)                      // pad_enable
              | (4u << 22)                      // pad_interval: 128B rows
              | (3u << 25));                    // pad_amount: 4 dwords
  g1[1] = (int)(256u << 16);                    // tensor_dim0 = 256 (lo16)
  g1[2] = (int)(4096u << 16);                   // dim0 hi=0 | tensor_dim1 = 4096
  g1[3] = (int)((u32)KCH << 16);                // dim1 hi=0 | tile_dim0 = 32
  g1[4] = 128;                                  // tile_dim1 = 128 rows
  g1[5] = 256;                                  // tensor_dim0_stride = 256
  g1[6] = 0;
  g1[7] = 0;
  v4i gz = {0, 0, 0, 0};
#if defined(__clang_major__) && (__clang_major__ >= 23)
  v8i gz8 = {0, 0, 0, 0, 0, 0, 0, 0};
  __builtin_amdgcn_tensor_load_to_lds(g0, g1, gz, gz, gz8, 0);
#else
  __builtin_amdgcn_tensor_load_to_lds(g0, g1, gz, gz, 0);
#endif
}

// ------ kernel 6: TDM-staged WMMA Gram (upper-tri 128x128) + fused loss ------
// 8 waves/block; wave w owns a 32x64 strip (2x4 sub-tiles, 8 f32 accumulators).
// Per k-step: 6 ds_load_b64 feed 8 v_wmma_f32_16x16x4_f32 (independent accs ->
// no RAW NOPs). LDS rows padded to 36 dwords: 64 dwords/ds_load_b64 hit all 64
// banks exactly once.
__global__ void __launch_bounds__(256)
gram_loss_kernel(const float* __restrict__ normed,
                 float* __restrict__ partials,
                 u32* __restrict__ nnz) {
  // Sole __shared__ object in this kernel -> LDS byte offset 0.
  __shared__ float smem[2 * PANEL_DW];
  int b = blockIdx.x, sm = 0;                   // map block -> (sm <= sn)
  while (b >= NS - sm) { b -= NS - sm; ++sm; }
  int sn = sm + b;
  int wave = threadIdx.y, lane = threadIdx.x;
  int hi = lane >> 4, lo = lane & 15;
  int rstrip = wave >> 1;                       // 0..3 (32-row strips)
  int cstrip = wave & 1;                        // 0..1 (64-col strips)

  v8f acc[8];
#pragma unroll
  for (int t = 0; t < 8; ++t) acc[t] = (v8f){0.f,0.f,0.f,0.f,0.f,0.f,0.f,0.f};

  int arow = rstrip * 32 + lo;                  // + sr*16
  int brow = PANEL_DW / LDS_STRIDE == 128 ? 0 : 0; (void)brow;
  int brow0 = cstrip * 64 + lo;                 // + sc*16

  for (int kc = 0; kc < OUT_CH; kc += KCH) {
    if (wave == 0) {
      tdm_stage_panel(normed, sm * 128, kc, 0u);
      tdm_stage_panel(normed, sn * 128, kc, (u32)PANEL_BYTES);
      __builtin_amdgcn_s_wait_tensorcnt(0);
    }
    __syncthreads();                            // panels visible to all waves
    // The compiler cannot see that the TDM wrote `smem` (descriptor-based
    // side effect). Escape the LDS pointer through a side-effecting asm with
    // a memory clobber so the fragment loads below are not folded to undef
    // and are not CSE'd across k-chunk restages. Emits zero instructions.
    asm volatile("" : : "v"(&smem[0]) : "memory");
#pragma unroll
    for (int k = 0; k < KCH; k += 4) {
      int kk = k + 2 * hi;
      v2f a0 = *(const v2f*)&smem[(arow +  0) * LDS_STRIDE + kk];
      v2f a1 = *(const v2f*)&smem[(arow + 16) * LDS_STRIDE + kk];
      v2f b0 = *(const v2f*)&smem[PANEL_DW + (brow0 +  0) * LDS_STRIDE + kk];
      v2f b1 = *(const v2f*)&smem[PANEL_DW + (brow0 + 16) * LDS_STRIDE + kk];
      v2f b2 = *(const v2f*)&smem[PANEL_DW + (brow0 + 32) * LDS_STRIDE + kk];
      v2f b3 = *(const v2f*)&smem[PANEL_DW + (brow0 + 48) * LDS_STRIDE + kk];
      acc[0] = __builtin_amdgcn_wmma_f32_16x16x4_f32(false, a0, false, b0, (short)0, acc[0], false, false);
      acc[1] = __builtin_amdgcn_wmma_f32_16x16x4_f32(false, a1, false, b0, (short)0, acc[1], false, false);
      acc[2] = __builtin_amdgcn_wmma_f32_16x16x4_f32(false, a0, false, b1, (short)0, acc[2], false, false);
      acc[3] = __builtin_amdgcn_wmma_f32_16x16x4_f32(false, a1, false, b1, (short)0, acc[3], false, false);
      acc[4] = __builtin_amdgcn_wmma_f32_16x16x4_f32(false, a0, false, b2, (short)0, acc[4], false, false);
      acc[5] = __builtin_amdgcn_wmma_f32_16x16x4_f32(false, a1, false, b2, (short)0, acc[5], false, false);
      acc[6] = __builtin_amdgcn_wmma_f32_16x16x4_f32(false, a0, false, b3, (short)0, acc[6], false, false);
      acc[7] = __builtin_amdgcn_wmma_f32_16x16x4_f32(false, a1, false, b3, (short)0, acc[7], false, false);
    }
    __syncthreads();                            // all reads done before restage
  }

  // fused contrastive-loss epilogue over 8 sub-tiles (16x16 each)
  float sum = 0.0f;
  u32 cnt = 0;
  int tm0 = (sm * 128 + rstrip * 32) >> 4;
  int tn0 = (sn * 128 + cstrip * 64) >> 4;
#pragma unroll
  for (int sc = 0; sc < 4; ++sc) {
#pragma unroll
    for (int sr = 0; sr < 2; ++sr) {
      int tm16 = tm0 + sr, tn16 = tn0 + sc;
      if (tm16 > tn16) continue;                // lower tri handled by symmetry
      float wgt = (tm16 == tn16) ? 1.0f : 2.0f;
      u32 cinc = (tm16 == tn16) ? 1u : 2u;
      v8f a = acc[sc * 2 + sr];
#pragma unroll
      for (int v = 0; v < 8; ++v) {
        int i = tm16 * 16 + v + 8 * hi;         // C layout: m = v + 8*(lane>=16)
        int j = tn16 * 16 + lo;                 // n = lane & 15
        float s = a[v] * TAU_INV;
        bool same = (i < P_NUM) == (j < P_NUM);
        float t = fmaxf(MARGIN - s, 0.0f);
        float val = same ? (t * t) : (s * s);
        if (i == j) val = 0.0f;
        if (val != 0.0f) { sum += wgt * val; cnt += cinc; }
      }
    }
  }
#pragma unroll
  for (int m = 16; m >= 1; m >>= 1) {
    sum += __shfl_xor(sum, m, 32);
    cnt += __shfl_xor(cnt, m, 32);
  }
  if (lane == 0) {
    partials[blockIdx.x * 8 + wave] = sum;      // unique slot -> deterministic
    atomicAdd(nnz, cnt);                        // integer add: exact
  }
}

// ---------------- kernel 7: deterministic final reduction ----------------
__global__ void final_kernel(const float* __restrict__ partials,
                             const u32* __restrict__ nnz,
                             float* __restrict__ out) {
  __shared__ double red[256];
  double acc = 0.0;
  for (int i = threadIdx.x; i < NPART; i += 256) acc += (double)partials[i];
  red[threadIdx.x] = acc;
  __syncthreads();
  for (int s = 128; s > 0; s >>= 1) {
    if (threadIdx.x < s) red[threadIdx.x] += red[threadIdx.x + s];
    __syncthreads();
  }
  if (threadIdx.x == 0) out[0] = (float)(red[0] / (double)(*nnz));
}

// ---------------- host-side launcher ----------------
extern "C" void kernel_launch(void* const* d_in, const int* in_sizes, int n_in,
                              void* d_out, int out_size, void* d_ws, size_t ws_size,
                              hipStream_t stream) {
  const float* d1    = (const float*)d_in[0];
  const float* pfegc = (const float*)d_in[1];
  const float* ppre  = (const float*)d_in[2];
  const float* gt    = (const float*)d_in[3];
  const float* Wfc   = (const float*)d_in[4];
  const float* bfc   = (const float*)d_in[5];
  float* out = (float*)d_out;
  char* ws = (char*)d_ws;

  u32* scoreP  = (u32*)(ws + OFF_SCOREP);
  u32* scoreN  = (u32*)(ws + OFF_SCOREN);
  u32* histP   = (u32*)(ws + OFF_HISTP);
  u32* histN   = (u32*)(ws + OFF_HISTN);
  u32* ctrs    = (u32*)(ws + OFF_CTRS);
  float* part  = (float*)(ws + OFF_PART);
  u32* thr     = (u32*)(ws + OFF_THR);
  u32* selP    = (u32*)(ws + OFF_SELP);
  u32* selN    = (u32*)(ws + OFF_SELN);
  u64* candP   = (u64*)(ws + OFF_CANDP);
  u64* candN   = (u64*)(ws + OFF_CANDN);
  float* normd = (float*)(ws + OFF_NORMED);

  hipMemsetAsync(ws + OFF_ZERO, 0, ZERO_BYTES, stream);

  score_kernel<<<HALF_N / 256, 256, 0, stream>>>(pfegc, ppre, gt,
                                                 scoreP, scoreN, histP, histN);
  threshold_kernel<<<1, 32, 0, stream>>>(histP, histN, thr);
  compact_kernel<<<NPIX / 256, 256, 0, stream>>>(scoreP, scoreN, thr, ctrs,
                                                 selP, selN, candP, candN);
  select_kernel<<<1, 256, 0, stream>>>(thr, ctrs, candP, candN, selP, selN);
  project_kernel<<<T_PTS, 256, 0, stream>>>(d1, Wfc, bfc, selP, selN, normd);

  dim3 gblk(32, 8);
  gram_loss_kernel<<<NPAIRS, gblk, 0, stream>>>(normd, part, &ctrs[4]);

  final_kernel<<<1, 256, 0, stream>>>(part, &ctrs[4], out);
}